// DynamicSequenceChunker_30468497998163
// MI455X (gfx1250) — compile-verified
//
#include <hip/hip_runtime.h>
#include <hip/hip_bf16.h>
#include <math.h>

#define Bx 8
#define Lx 4096
#define Dx 1024
#define NQK 2048
#define EPSV 1e-8f

typedef float v2f __attribute__((ext_vector_type(2)));
typedef float v8f __attribute__((ext_vector_type(8)));

// workspace layout (float32 / int32 units)
#define OFF_SN2   0
#define OFF_DOT0  8
#define OFF_AUX   16
#define OFF_NUM   24                    // int32[8]
#define OFF_PROBS 64
#define OFF_BPOS  (64 + Bx*Lx)          // int32[B*L]
#define OFF_BPROB (64 + 2*Bx*Lx)        // float[B*L]

// ---------------------------------------------------------------------------
// Kernel 1: v = Wq @ start_key ; sn2 = |s|^2 ; dot0[b] = tokens[b,0,:] . v
// ---------------------------------------------------------------------------
__global__ __launch_bounds__(1024)
void dsc_prep(const float* __restrict__ W, const float* __restrict__ skey,
              const float* __restrict__ tokens, float* __restrict__ ws) {
  __shared__ float vsh[1024];
  __shared__ float red[1024];
  const int d = threadIdx.x;

  float acc = 0.f;
  const float* wr = W + (size_t)d * NQK;   // row d of W_qk, query half = cols [0,1024)
  for (int n = 0; n < Dx; ++n) acc += wr[n] * skey[n];
  vsh[d] = acc;

  const float sv = skey[d];
  red[d] = sv * sv;
  __syncthreads();
  for (int off = 512; off > 0; off >>= 1) {
    if (d < off) red[d] += red[d + off];
    __syncthreads();
  }
  if (d == 0) ws[OFF_SN2] = red[0];

  for (int b = 0; b < Bx; ++b) {
    __syncthreads();
    red[d] = tokens[((size_t)b * Lx) * Dx + d] * vsh[d];
    __syncthreads();
    for (int off = 512; off > 0; off >>= 1) {
      if (d < off) red[d] += red[d + off];
      __syncthreads();
    }
    if (d == 0) ws[OFF_DOT0 + b] = red[0];
  }
}

// ---------------------------------------------------------------------------
// Kernel 2: fused fp32 WMMA GEMM + cosine statistics -> probs[b,l]
//   Each block: 16 token rows, FULL-WIDTH tiles staged once in LDS (131.6 KB,
//   CDNA5 allows 320 KB/WGP). T0 = rows l0..l0+15 (queries), T1 = rows
//   l0-1..l0+14 (shifted keys). 8 waves free-run over 64 N-chunks, no inner
//   barriers.
// ---------------------------------------------------------------------------
#define TSTRIDE 1028   // pad: bank = (m*4 + k) % 64 -> conflict-free; float2 aligned

__global__ __launch_bounds__(256)
void dsc_gemm_probs(const float* __restrict__ tokens, const float* __restrict__ W,
                    float* __restrict__ ws) {
  __shared__ float T0[16 * TSTRIDE];
  __shared__ float T1[16 * TSTRIDE];
  __shared__ float part_q[8][16];
  __shared__ float part_k[8][16];
  __shared__ float part_d[8][16];

  const int tid  = threadIdx.x;
  const int wave = tid >> 5;
  const int lane = tid & 31;
  const int b    = blockIdx.x >> 8;          // 256 row-tiles per batch
  const int l0   = (blockIdx.x & 255) << 4;

  const int m    = lane & 15;                // A-frag row (M)
  const int koff = (lane >> 4) << 1;         // lanes 16-31 hold K+2,K+3
  const int ncol = lane & 15;                // B-frag column (N)

  const float* tokB = tokens + ((size_t)b * Lx) * Dx;

  // ---- stage full 16x1024 tiles once (4096 float4 per tile, 16 per thread) --
#pragma unroll
  for (int it = 0; it < 16; ++it) {
    const int f   = tid + it * 256;          // float4 index 0..4095
    const int row = f >> 8;                  // 256 float4 per row
    const int c4  = (f & 255) << 2;
    const float4 v0 = *(const float4*)&tokB[((size_t)(l0 + row)) * Dx + c4];
    *(float4*)&T0[row * TSTRIDE + c4] = v0;
    int g1 = l0 + row - 1; if (g1 < 0) g1 = 0;     // l=0 fixed via start_key later
    const float4 v1 = *(const float4*)&tokB[((size_t)g1) * Dx + c4];
    *(float4*)&T1[row * TSTRIDE + c4] = v1;
  }
  __syncthreads();

  float acc_q[8], acc_k[8], acc_d[8];
#pragma unroll
  for (int j = 0; j < 8; ++j) { acc_q[j] = 0.f; acc_k[j] = 0.f; acc_d[j] = 0.f; }

  const float* a0base = &T0[m * TSTRIDE + koff];
  const float* a1base = &T1[m * TSTRIDE + koff];

  for (int cc = 0; cc < 8; ++cc) {
    const int nc = cc * 8 + wave;            // N-chunk in [0,64)
    const int n0 = nc << 4;
    const float* wq = W + (size_t)(n0 + ncol);        // query cols [0,1024)
    const float* wk = W + (size_t)(Dx + n0 + ncol);   // key   cols [1024,2048)

    v8f cq = {0.f, 0.f, 0.f, 0.f, 0.f, 0.f, 0.f, 0.f};
    v8f ck = {0.f, 0.f, 0.f, 0.f, 0.f, 0.f, 0.f, 0.f};

#pragma unroll 4
    for (int k0 = 0; k0 < Dx; k0 += 4) {
      v2f aq = *(const v2f*)&a0base[k0];      // ds_load (float2, conflict-free)
      v2f ak = *(const v2f*)&a1base[k0];
      const size_t krow = (size_t)(k0 + koff) * NQK;
      v2f bq = { wq[krow], wq[krow + NQK] };
      v2f bk = { wk[krow], wk[krow + NQK] };
      cq = __builtin_amdgcn_wmma_f32_16x16x4_f32(false, aq, false, bq,
                                                 (short)0, cq, false, false);
      ck = __builtin_amdgcn_wmma_f32_16x16x4_f32(false, ak, false, bk,
                                                 (short)0, ck, false, false);
    }
#pragma unroll
    for (int j = 0; j < 8; ++j) {
      acc_q[j] += cq[j] * cq[j];   // |q|^2 partial over this N-chunk
      acc_k[j] += ck[j] * ck[j];   // |k_shift|^2 partial
      acc_d[j] += cq[j] * ck[j];   // q . k_shift partial
    }
  }

  // reduce across the 16 lanes that share a row (N direction)
#pragma unroll
  for (int j = 0; j < 8; ++j) {
    float vq = acc_q[j], vk = acc_k[j], vd = acc_d[j];
    for (int off = 8; off >= 1; off >>= 1) {
      vq += __shfl_xor(vq, off, 16);
      vk += __shfl_xor(vk, off, 16);
      vd += __shfl_xor(vd, off, 16);
    }
    if ((lane & 15) == 0) {
      const int row = j + ((lane >> 4) << 3);   // C layout: M = j + 8*(lane>=16)
      part_q[wave][row] = vq;
      part_k[wave][row] = vk;
      part_d[wave][row] = vd;
    }
  }
  __syncthreads();

  if (tid < 16) {                 // deterministic fixed-order combine (no fp atomics)
    float q2 = 0.f, k2 = 0.f, dt = 0.f;
#pragma unroll
    for (int w = 0; w < 8; ++w) { q2 += part_q[w][tid]; k2 += part_k[w][tid]; dt += part_d[w][tid]; }
    const int l = l0 + tid;
    float kn = sqrtf(k2);
    if (l == 0) { dt = ws[OFF_DOT0 + b]; kn = sqrtf(ws[OFF_SN2]); }
    const float qn = fmaxf(sqrtf(q2), EPSV);
    kn = fmaxf(kn, EPSV);
    const float cs = dt / (qn * kn);
    ws[OFF_PROBS + (size_t)b * Lx + l] = (1.f - cs) * 0.5f;
  }
}

// ---------------------------------------------------------------------------
// Kernel 3: per-batch boundary scan, compaction, chunk_lens, aux partial
// ---------------------------------------------------------------------------
__global__ __launch_bounds__(1024)
void dsc_scan(float* __restrict__ ws, float* __restrict__ lens_out) {
  const int b = blockIdx.x;
  const int t = threadIdx.x;
  __shared__ int   s0[1024];
  __shared__ int   s1[1024];
  __shared__ float fred[1024];
  __shared__ int   bpos_s[Lx];

  const float* probs = ws + OFF_PROBS + (size_t)b * Lx;
  int*   bpos  = (int*)ws + OFF_BPOS + b * Lx;
  float* bprob = ws + OFF_BPROB + (size_t)b * Lx;
  int*   numc  = (int*)ws + OFF_NUM;

  float p[4]; int mk[4]; int lsum = 0;
#pragma unroll
  for (int i = 0; i < 4; ++i) {
    const int pos = t * 4 + i;
    p[i]  = probs[pos];
    mk[i] = (p[i] > 0.5f) || (pos == 0);   // first token forced boundary
    lsum += mk[i];
  }
  s0[t] = lsum;
  __syncthreads();

  int* src = s0; int* dst = s1;
  for (int off = 1; off < 1024; off <<= 1) {   // Hillis-Steele inclusive scan
    int v = src[t];
    if (t >= off) v += src[t - off];
    dst[t] = v;
    __syncthreads();
    int* tmp = src; src = dst; dst = tmp;
  }
  const int incl  = src[t];
  const int total = src[1023];
  int r = incl - lsum;                         // exclusive base
#pragma unroll
  for (int i = 0; i < 4; ++i) {
    if (mk[i]) { bpos_s[r] = t * 4 + i; bpos[r] = t * 4 + i; bprob[r] = p[i]; ++r; }
  }
  if (t == 0) numc[b] = total;
  __syncthreads();

#pragma unroll
  for (int i = 0; i < 4; ++i) {
    const int rr = t * 4 + i;
    float len = 0.f;
    if (rr < total) {
      const int pe = (rr + 1 < total) ? bpos_s[rr + 1] : Lx;   // end sentinel = L
      len = (float)(pe - bpos_s[rr]);
    }
    lens_out[(size_t)b * Lx + rr] = len;
  }

  fred[t] = p[0] + p[1] + p[2] + p[3];
  __syncthreads();
  for (int off = 512; off > 0; off >>= 1) {
    if (t < off) fred[t] += fred[t + off];
    __syncthreads();
  }
  if (t == 0) {
    const float Nt = 6.f;
    const float G  = fred[0] / (float)Lx;
    const float Fm = (float)total / (float)Lx;   // forward value of F is F_hard
    ws[OFF_AUX + b] = Nt / (Nt - 1.f) * ((Nt - 1.f) * Fm * G + (1.f - Fm) * (1.f - G));
  }
}

// ---------------------------------------------------------------------------
// Kernel 4: downsampled[b,r,:] = tokens[b,bpos,:] * bprob (zeros for pads)
// ---------------------------------------------------------------------------
__global__ __launch_bounds__(256)
void dsc_gather(const float* __restrict__ tokens, const float* __restrict__ ws,
                float* __restrict__ out) {
  const int row = blockIdx.x;                 // b*L + r
  const int b = row >> 12;
  const int r = row & (Lx - 1);
  const int t = threadIdx.x;
  const int*   numc  = (const int*)ws + OFF_NUM;
  const int*   bpos  = (const int*)ws + OFF_BPOS;
  const float* bprob = ws + OFF_BPROB;

  float4 v = {0.f, 0.f, 0.f, 0.f};
  if (r < numc[b]) {
    const int   p  = bpos[(size_t)b * Lx + r];
    const float sc = bprob[(size_t)b * Lx + r];
    const float4 tv = *(const float4*)&tokens[(((size_t)b * Lx) + p) * Dx + t * 4];
    v.x = tv.x * sc; v.y = tv.y * sc; v.z = tv.z * sc; v.w = tv.w * sc;
  }
  *(float4*)&out[((size_t)row) * Dx + t * 4] = v;
}

// ---------------------------------------------------------------------------
// Kernel 5: scalar loss (fixed-order sum -> deterministic)
// ---------------------------------------------------------------------------
__global__ void dsc_loss(const float* __restrict__ ws, float* __restrict__ out) {
  if (threadIdx.x == 0 && blockIdx.x == 0) {
    float s = 0.f;
    for (int b = 0; b < Bx; ++b) s += ws[OFF_AUX + b];
    out[0] = (s / (float)Bx) * 0.03f;
  }
}

extern "C" void kernel_launch(void* const* d_in, const int* in_sizes, int n_in,
                              void* d_out, int out_size, void* d_ws, size_t ws_size,
                              hipStream_t stream) {
  const float* tokens = (const float*)d_in[0];   // [8,4096,1024]
  const float* W      = (const float*)d_in[1];   // [1024,2048]
  const float* skey   = (const float*)d_in[2];   // [1024]
  float* out = (float*)d_out;
  float* ws  = (float*)d_ws;

  float* dsout = out;                                   // B*L*D
  float* lens  = out + (size_t)Bx * Lx * Dx;            // B*L (as f32)
  float* loss  = out + (size_t)Bx * Lx * Dx + Bx * Lx;  // 1

  hipLaunchKernelGGL(dsc_prep,       dim3(1),            dim3(1024), 0, stream, W, skey, tokens, ws);
  hipLaunchKernelGGL(dsc_gemm_probs, dim3(Bx * Lx / 16), dim3(256),  0, stream, tokens, W, ws);
  hipLaunchKernelGGL(dsc_scan,       dim3(Bx),           dim3(1024), 0, stream, ws, lens);
  hipLaunchKernelGGL(dsc_gather,     dim3(Bx * Lx),      dim3(256),  0, stream, tokens, ws, dsout);
  hipLaunchKernelGGL(dsc_loss,       dim3(1),            dim3(32),   0, stream, ws, loss);
}